// GTLayer_10565619548708
// MI455X (gfx1250) — compile-verified
//
#include <hip/hip_runtime.h>

#define D 128
#define HEAD 4
#define NT 8                 // number of 16-wide N tiles per wave (D/16)

typedef __attribute__((ext_vector_type(2))) float v2f;
typedef __attribute__((ext_vector_type(8))) float v8f;

// ---------------------------------------------------------------------------
// Kernel 1: zero the scatter-add targets (output embeds + attNorm in ws).
// ---------------------------------------------------------------------------
__global__ void gt_zero_init(float* __restrict__ outEmb,
                             float* __restrict__ attNorm,
                             int nOut, int nNorm) {
    int i = blockIdx.x * blockDim.x + threadIdx.x;
    int stride = gridDim.x * blockDim.x;
    for (int j = i; j < nOut; j += stride) outEmb[j] = 0.0f;
    for (int j = i; j < nNorm; j += stride) attNorm[j] = 0.0f;
}

// ---------------------------------------------------------------------------
// Kernel 2: Q/K/V = embeds @ W via V_WMMA_F32_16X16X4_F32 (wave32).
// One wave computes a full 16x128 output strip (8 N-tiles), so each A
// fragment is loaded once and reused by 8 consecutive WMMAs. grid.y selects
// which weight matrix (q/k/v).
//
// VGPR layouts per CDNA5 ISA 7.12.2:
//  A (16x4):  lanes 0-15 -> M=lane,   VGPR0=K0, VGPR1=K1
//             lanes16-31 -> M=lane-16,VGPR0=K2, VGPR1=K3
//  B (4x16):  lanes 0-15 -> N=lane,   VGPR0=K0, VGPR1=K1
//             lanes16-31 -> N=lane-16,VGPR0=K2, VGPR1=K3
//  C/D (16x16, 8 VGPRs): lane<16 -> M=g, lane>=16 -> M=g+8, N=lane&15
// ---------------------------------------------------------------------------
__global__ void gt_qkv_wmma(const float* __restrict__ embeds,
                            const float* __restrict__ qW,
                            const float* __restrict__ kW,
                            const float* __restrict__ vW,
                            float* __restrict__ Q,
                            float* __restrict__ K,
                            float* __restrict__ V,
                            int nNodes) {
    const int lane = threadIdx.x;          // 0..31 (wave32 block)
    const int half = lane >> 4;            // 0: lanes 0-15, 1: lanes 16-31
    const int l15  = lane & 15;
    const int tileM = blockIdx.x;

    const float* W;
    float* O;
    if (blockIdx.y == 0)      { W = qW; O = Q; }
    else if (blockIdx.y == 1) { W = kW; O = K; }
    else                      { W = vW; O = V; }

    // Clamp row for loads so the last partial tile stays in-bounds
    // (stores are masked below); keeps EXEC all-1s for the WMMA.
    int m  = tileM * 16 + l15;
    int mc = (m < nNodes) ? m : (nNodes - 1);
    const float* arow = embeds + (size_t)mc * D;

    v8f acc[NT];
#pragma unroll
    for (int t = 0; t < NT; ++t) acc[t] = (v8f){};

#pragma unroll 2
    for (int kk = 0; kk < D; kk += 4) {
        const int ka = kk + (half << 1);   // K index for VGPR0 of this lane half
        v2f a;
        a.x = arow[ka];
        a.y = arow[ka + 1];
        const float* wr0 = W + (size_t)ka * D + l15;        // row ka   (VGPR0)
        const float* wr1 = wr0 + D;                         // row ka+1 (VGPR1)
#pragma unroll
        for (int t = 0; t < NT; ++t) {
            v2f b;
            b.x = wr0[t * 16];
            b.y = wr1[t * 16];
            // (neg_a, A, neg_b, B, c_mod, C, reuse_a, reuse_b)
            acc[t] = __builtin_amdgcn_wmma_f32_16x16x4_f32(
                false, a, false, b, (short)0, acc[t], false, false);
        }
    }

#pragma unroll
    for (int t = 0; t < NT; ++t) {
#pragma unroll
        for (int g = 0; g < 8; ++g) {
            int row = tileM * 16 + g + half * 8;
            if (row < nNodes)
                O[(size_t)row * D + t * 16 + l15] = acc[t][g];
        }
    }
}

// ---------------------------------------------------------------------------
// Kernel 3: per-edge attention scores. One wave32 per edge per iteration.
// Lane L covers dims [4L, 4L+4) (float4); head = L/8; 8-lane xor-shuffle
// reduction yields per-head dot(Q[row], K[col]); clip, exp, atomic norm.
// ---------------------------------------------------------------------------
__global__ void gt_att_score(const float* __restrict__ Q,
                             const float* __restrict__ K,
                             const int* __restrict__ rows,
                             const int* __restrict__ cols,
                             float* __restrict__ expAtt,
                             float* __restrict__ attNorm,
                             int E) {
    const int lane = threadIdx.x & 31;
    const int wid  = (blockIdx.x * blockDim.x + threadIdx.x) >> 5;
    const int nw   = (gridDim.x * blockDim.x) >> 5;

    for (int e = wid; e < E; e += nw) {
        const int r = rows[e];
        const int c = cols[e];
        const float4 q = *reinterpret_cast<const float4*>(Q + (size_t)r * D + lane * 4);
        const float4 k = *reinterpret_cast<const float4*>(K + (size_t)c * D + lane * 4);
        float p = q.x * k.x + q.y * k.y + q.z * k.z + q.w * k.w;
        // reduce within each 8-lane head group (wave32 shuffles)
        p += __shfl_xor(p, 1, 32);
        p += __shfl_xor(p, 2, 32);
        p += __shfl_xor(p, 4, 32);
        if ((lane & 7) == 0) {
            const int h = lane >> 3;
            float a  = fminf(fmaxf(p, -10.0f), 10.0f);
            float ea = __expf(a);
            expAtt[(size_t)e * HEAD + h] = ea;
            atomicAdd(attNorm + (size_t)r * HEAD + h, ea);
        }
    }
}

// ---------------------------------------------------------------------------
// Kernel 4: normalize attention, emit att output, scatter-add att*V[col]
// into resEmbeds[row].
// ---------------------------------------------------------------------------
__global__ void gt_aggregate(const float* __restrict__ V,
                             const float* __restrict__ expAtt,
                             const float* __restrict__ attNorm,
                             const int* __restrict__ rows,
                             const int* __restrict__ cols,
                             float* __restrict__ outEmb,
                             float* __restrict__ outAtt,
                             int E) {
    const int lane = threadIdx.x & 31;
    const int wid  = (blockIdx.x * blockDim.x + threadIdx.x) >> 5;
    const int nw   = (gridDim.x * blockDim.x) >> 5;

    for (int e = wid; e < E; e += nw) {
        const int r = rows[e];
        const int c = cols[e];
        const int h = lane >> 3;
        const float ea   = expAtt[(size_t)e * HEAD + h];
        const float norm = attNorm[(size_t)r * HEAD + h];
        const float a = ea / (norm + 1e-8f);
        if ((lane & 7) == 0)
            outAtt[(size_t)e * HEAD + h] = a;
        const float4 v = *reinterpret_cast<const float4*>(V + (size_t)c * D + lane * 4);
        float* dst = outEmb + (size_t)r * D + lane * 4;
        atomicAdd(dst + 0, a * v.x);
        atomicAdd(dst + 1, a * v.y);
        atomicAdd(dst + 2, a * v.z);
        atomicAdd(dst + 3, a * v.w);
    }
}

// ---------------------------------------------------------------------------
extern "C" void kernel_launch(void* const* d_in, const int* in_sizes, int n_in,
                              void* d_out, int out_size, void* d_ws, size_t ws_size,
                              hipStream_t stream) {
    const float* embeds = (const float*)d_in[0];
    const float* qW     = (const float*)d_in[1];
    const float* kW     = (const float*)d_in[2];
    const float* vW     = (const float*)d_in[3];
    const int*   rows   = (const int*)d_in[4];
    const int*   cols   = (const int*)d_in[5];

    const int N = in_sizes[0] / D;   // 25000
    const int E = in_sizes[4];       // 400000

    float* out    = (float*)d_out;
    float* outEmb = out;                          // [N, D]
    float* outAtt = out + (size_t)N * D;          // [E, HEAD]

    // Workspace layout: Q | K | V | attNorm | expAtt  (~45.2 MB)
    float* Q       = (float*)d_ws;
    float* K       = Q + (size_t)N * D;
    float* V       = K + (size_t)N * D;
    float* attNorm = V + (size_t)N * D;
    float* expAtt  = attNorm + (size_t)N * HEAD;

    gt_zero_init<<<512, 256, 0, stream>>>(outEmb, attNorm, N * D, N * HEAD);

    dim3 gGemm((N + 15) / 16, 3);           // 1563 x 3 waves, 16x128 strip each
    gt_qkv_wmma<<<gGemm, 32, 0, stream>>>(embeds, qW, kW, vW, Q, K, V, N);

    gt_att_score<<<2048, 256, 0, stream>>>(Q, K, rows, cols, expAtt, attNorm, E);

    gt_aggregate<<<2048, 256, 0, stream>>>(V, expAtt, attNorm, rows, cols,
                                           outEmb, outAtt, E);
}